// Head_58686433132809
// MI455X (gfx1250) — compile-verified
//
#include <hip/hip_runtime.h>
#include <hip/hip_bf16.h>
#include <stdint.h>

typedef __attribute__((ext_vector_type(2))) float v2f;
typedef __attribute__((ext_vector_type(4))) float v4f;
typedef __attribute__((ext_vector_type(8))) float v8f;
typedef __attribute__((ext_vector_type(4))) unsigned int u32x4;
typedef __attribute__((ext_vector_type(4))) int          i32x4;
typedef __attribute__((ext_vector_type(8))) int          i32x8;

#define HEAD  64
#define CDIM  768
#define TSEQ  2048
#define BATCH 8
#define NROWS (BATCH * TSEQ)   // 16384
#define KROW  68               // padded LDS row stride (floats): conflict-free fragments

// --- TDM availability / arity detection (bridge doc: 5-arg ROCm7.2, 6-arg therock) ---
#if defined(__AMDGCN__) && __has_builtin(__builtin_amdgcn_tensor_load_to_lds)
#define HAVE_TDM 1
#if __has_include(<hip/amd_detail/amd_gfx1250_TDM.h>)
#define TDM_ARGS6 1
#endif
#endif

// ---------------------------------------------------------------------------
// Kernel 1: QKV projection (unchanged from round 1 — WMMA-dense, memory bound).
// One wave computes one 16x16 fp32 tile of Q/K/V via V_WMMA_F32_16X16X4_F32.
// ---------------------------------------------------------------------------
__global__ __launch_bounds__(128) void qkv_proj(
    const float* __restrict__ x,
    const float* __restrict__ Wq, const float* __restrict__ Wk,
    const float* __restrict__ Wv,
    float* __restrict__ Q, float* __restrict__ K, float* __restrict__ V)
{
  const int wave = threadIdx.x >> 5;
  const int lane = threadIdx.x & 31;
  const int r    = lane & 15;
  const int half = lane >> 4;
  const int row0 = blockIdx.x * 16;
  const int n0   = wave * 16;

  const float* W;  float* O;
  if (blockIdx.y == 0)      { W = Wq; O = Q; }
  else if (blockIdx.y == 1) { W = Wk; O = K; }
  else                      { W = Wv; O = V; }

  const float* xrow = x + (size_t)(row0 + r) * CDIM + 2 * half;

  v8f acc = {};
  for (int c0 = 0; c0 < CDIM; c0 += 4) {
    v2f a = *(const v2f*)(xrow + c0);
    const float* wp = W + (size_t)(c0 + 2 * half) * HEAD + n0 + r;
    v2f b;
    b.x = wp[0];
    b.y = wp[HEAD];
    acc = __builtin_amdgcn_wmma_f32_16x16x4_f32(
        false, a, false, b, (short)0, acc, false, false);
  }

  float* orow = O + (size_t)(row0 + 8 * half) * HEAD + n0 + r;
#pragma unroll
  for (int i = 0; i < 8; ++i)
    orow[(size_t)i * HEAD] = acc[i];
}

// ---------------------------------------------------------------------------
// TDM: DMA one 16x64 fp32 tile (row stride 64 floats in memory) into LDS with
// 4 dwords of padding inserted every 64 dwords -> 68-float LDS row stride.
// D# packing per CDNA5 ISA §8.3/§8.4.
// ---------------------------------------------------------------------------
#ifdef HAVE_TDM
__device__ __forceinline__ void tdm_load_tile16x64(const float* gsrc,
                                                   unsigned int lds_byte_off)
{
  const unsigned long long ga = (unsigned long long)(uintptr_t)gsrc;

  u32x4 g0;
  g0.x = 1u;                                             // count=1, user desc
  g0.y = lds_byte_off;                                   // lds_addr (bytes)
  g0.z = (unsigned int)ga;                               // global_addr[31:0]
  g0.w = (unsigned int)((ga >> 32) & 0x01FFFFFFu)        // global_addr[56:32]
       | (2u << 30);                                     // type=2 ("image")

  i32x8 g1;
  g1[0] = (int)((2u << 16)        // data_size = 4 bytes
              | (1u << 20)        // pad_enable
              | (5u << 22)        // pad_interval: every 64 dwords
              | (3u << 25));      // pad_amount: 4 dwords
  g1[1] = (int)(64u << 16);       // tensor_dim0 = 64 elements
  g1[2] = (int)(16u << 16);       // tensor_dim1 = 16 rows
  g1[3] = (int)(64u << 16);       // tile_dim0 = 64
  g1[4] = 16;                     // tile_dim1 = 16, tile_dim2 = 0
  g1[5] = 64;                     // tensor_dim0_stride = 64 elements
  g1[6] = 0;
  g1[7] = 0;

  i32x4 z4 = {0, 0, 0, 0};
#ifdef TDM_ARGS6
  i32x8 z8 = {0, 0, 0, 0, 0, 0, 0, 0};
  __builtin_amdgcn_tensor_load_to_lds(g0, g1, z4, z4, z8, 0);
#else
  __builtin_amdgcn_tensor_load_to_lds(g0, g1, z4, z4, 0);
#endif
}
#endif

// ---------------------------------------------------------------------------
// Kernel 2: flash attention. Block = 4 waves = 4 consecutive query tiles of
// ONE batch, sharing double-buffered K/V tiles staged in LDS by the TDM
// (wave0 -> K tile, wave1 -> V tile), overlapped with WMMA compute.
// ---------------------------------------------------------------------------
__global__ __launch_bounds__(128) void attn(
    const float* __restrict__ Q, const float* __restrict__ K,
    const float* __restrict__ V, float* __restrict__ out)
{
  __shared__ float kbuf[2][16 * KROW];
  __shared__ float vbuf[2][16 * KROW];
  __shared__ float plds[4][16 * 18];

  const int wave = threadIdx.x >> 5;
  const int lane = threadIdx.x & 31;
  const int r    = lane & 15;
  const int half = lane >> 4;
  const int b    = blockIdx.x >> 5;                       // 32 blocks per batch
  const int q0   = ((blockIdx.x & 31) * 4 + wave) * 16;

  const float* Qb = Q + (size_t)b * TSEQ * HEAD;
  const float* Kb = K + (size_t)b * TSEQ * HEAD;
  const float* Vb = V + (size_t)b * TSEQ * HEAD;

  // Stage tile kt_ of K and V into LDS buffer buf_.
  auto fill = [&](int kt_, int buf_) {
#ifdef HAVE_TDM
    if (wave == 0)
      tdm_load_tile16x64(Kb + (size_t)(kt_ * 16) * HEAD,
                         (unsigned int)(uintptr_t)&kbuf[buf_][0]);
    else if (wave == 1)
      tdm_load_tile16x64(Vb + (size_t)(kt_ * 16) * HEAD,
                         (unsigned int)(uintptr_t)&vbuf[buf_][0]);
#else
    const float* gk = Kb + (size_t)(kt_ * 16) * HEAD;
    const float* gv = Vb + (size_t)(kt_ * 16) * HEAD;
#pragma unroll
    for (int it = 0; it < 2; ++it) {
      const int t   = it * 128 + (int)threadIdx.x;        // 256 float4 chunks
      const int row = t >> 4;
      const int ch  = (t & 15) * 4;
      *(v4f*)&kbuf[buf_][row * KROW + ch] = *(const v4f*)(gk + row * HEAD + ch);
      *(v4f*)&vbuf[buf_][row * KROW + ch] = *(const v4f*)(gv + row * HEAD + ch);
    }
#endif
  };

  // Preload Q tile as 16 A-fragments (K-slices of 4 head dims).
  v2f qa[16];
  {
    const float* qrow = Qb + (size_t)(q0 + r) * HEAD + 2 * half;
#pragma unroll
    for (int c = 0; c < 16; ++c)
      qa[c] = *(const v2f*)(qrow + c * 4);
  }

  v8f o0 = {}, o1 = {}, o2 = {}, o3 = {};
  float m[8], l[8];
#pragma unroll
  for (int i = 0; i < 8; ++i) { m[i] = -__builtin_inff(); l[i] = 0.f; }

  float* pl = plds[wave];
  const float scale = 0.125f;          // 1/sqrt(64)

  fill(0, 0);                          // prologue DMA of tile 0

  for (int kt = 0; kt < TSEQ / 16; ++kt) {
    const int cur = kt & 1;
#ifdef HAVE_TDM
    if (wave < 2) __builtin_amdgcn_s_wait_tensorcnt(0);  // my DMA of `cur` done
#endif
    __syncthreads();                   // cur visible to all; prev buffer free
    if (kt + 1 < TSEQ / 16) fill(kt + 1, cur ^ 1);       // overlap next DMA

    const float* kb = &kbuf[cur][0];
    const float* vb = &vbuf[cur][0];

    // ---- S = Q @ K^T  (16x16 fp32) : B fragments from LDS, stride KROW ----
    v8f s = {};
    const float* krow = kb + r * KROW + 2 * half;
#pragma unroll
    for (int c = 0; c < 16; ++c) {
      v2f bv = *(const v2f*)(krow + c * 4);
      s = __builtin_amdgcn_wmma_f32_16x16x4_f32(
          false, qa[c], false, bv, (short)0, s, false, false);
    }

    // ---- online softmax: row stats reduced across the 16-lane half ----
    float alpha[8];
#pragma unroll
    for (int i = 0; i < 8; ++i) {
      float sv = s[i] * scale;
      float rm = sv;
      rm = fmaxf(rm, __shfl_xor(rm, 1, 32));
      rm = fmaxf(rm, __shfl_xor(rm, 2, 32));
      rm = fmaxf(rm, __shfl_xor(rm, 4, 32));
      rm = fmaxf(rm, __shfl_xor(rm, 8, 32));
      const float mn = fmaxf(m[i], rm);
      const float p  = __expf(sv - mn);
      float rs = p;
      rs += __shfl_xor(rs, 1, 32);
      rs += __shfl_xor(rs, 2, 32);
      rs += __shfl_xor(rs, 4, 32);
      rs += __shfl_xor(rs, 8, 32);
      const float corr = __expf(m[i] - mn);
      l[i] = l[i] * corr + rs;
      m[i] = mn;
      alpha[i] = corr;
      pl[(i + 8 * half) * 18 + r] = p;           // D-layout -> row-major LDS
    }
#pragma unroll
    for (int i = 0; i < 8; ++i) {
      o0[i] *= alpha[i]; o1[i] *= alpha[i];
      o2[i] *= alpha[i]; o3[i] *= alpha[i];
    }

    // ---- O += P @ V : A from LDS (P transpose), B from staged V tile ----
#pragma unroll
    for (int kk = 0; kk < 16; kk += 4) {
      v2f pa = *(const v2f*)(pl + r * 18 + kk + 2 * half);
      const float* vrow = vb + (kk + 2 * half) * KROW;
      v2f b0, b1, b2, b3;
      b0.x = vrow[r];        b0.y = vrow[KROW + r];
      b1.x = vrow[16 + r];   b1.y = vrow[KROW + 16 + r];
      b2.x = vrow[32 + r];   b2.y = vrow[KROW + 32 + r];
      b3.x = vrow[48 + r];   b3.y = vrow[KROW + 48 + r];
      o0 = __builtin_amdgcn_wmma_f32_16x16x4_f32(false, pa, false, b0, (short)0, o0, false, false);
      o1 = __builtin_amdgcn_wmma_f32_16x16x4_f32(false, pa, false, b1, (short)0, o1, false, false);
      o2 = __builtin_amdgcn_wmma_f32_16x16x4_f32(false, pa, false, b2, (short)0, o2, false, false);
      o3 = __builtin_amdgcn_wmma_f32_16x16x4_f32(false, pa, false, b3, (short)0, o3, false, false);
    }
  }

  // ---- epilogue: normalize by l and store ----
  float* orow = out + ((size_t)b * TSEQ + q0 + 8 * half) * HEAD;
#pragma unroll
  for (int i = 0; i < 8; ++i) {
    const float inv = 1.0f / l[i];
    orow[(size_t)i * HEAD + r]      = o0[i] * inv;
    orow[(size_t)i * HEAD + 16 + r] = o1[i] * inv;
    orow[(size_t)i * HEAD + 32 + r] = o2[i] * inv;
    orow[(size_t)i * HEAD + 48 + r] = o3[i] * inv;
  }
}

// ---------------------------------------------------------------------------
extern "C" void kernel_launch(void* const* d_in, const int* in_sizes, int n_in,
                              void* d_out, int out_size, void* d_ws, size_t ws_size,
                              hipStream_t stream) {
  const float* x  = (const float*)d_in[0];
  const float* Wk = (const float*)d_in[1];
  const float* Wq = (const float*)d_in[2];
  const float* Wv = (const float*)d_in[3];
  float* out = (float*)d_out;

  float* Qp = (float*)d_ws;                    // 3 x 4 MB fp32 workspace
  float* Kp = Qp + (size_t)NROWS * HEAD;
  float* Vp = Kp + (size_t)NROWS * HEAD;

  dim3 gproj(NROWS / 16, 3);
  qkv_proj<<<gproj, 128, 0, stream>>>(x, Wq, Wk, Wv, Qp, Kp, Vp);

  attn<<<(NROWS / 16) / 4, 128, 0, stream>>>(Qp, Kp, Vp, out);
}